// TAGConvEncoder_10694468567648
// MI455X (gfx1250) — compile-verified
//
#include <hip/hip_runtime.h>

typedef float v2f __attribute__((ext_vector_type(2)));
typedef float v8f __attribute__((ext_vector_type(8)));

#define CH   128
#define KCAT 512   // (K_HOPS+1) * CH

// ---------------- graph norm precompute ----------------

__global__ void zero_f32(float* __restrict__ p, long n) {
  long i = (long)blockIdx.x * blockDim.x + threadIdx.x;
  if (i < n) p[i] = 0.0f;
}

__global__ void deg_kernel(const long long* __restrict__ ei, float* __restrict__ deg, int E) {
  int e = blockIdx.x * blockDim.x + threadIdx.x;
  if (e < E) {
    int col = (int)ei[(long)E + e];
    atomicAdd(deg + col, 1.0f);
  }
}

__global__ void dinv_kernel(float* __restrict__ deg, int n) {
  int i = blockIdx.x * blockDim.x + threadIdx.x;
  if (i < n) {
    float d = deg[i];
    deg[i] = (d > 0.0f) ? rsqrtf(d) : 0.0f;
  }
}

__global__ void norm_kernel(const long long* __restrict__ ei, const float* __restrict__ dinv,
                            float* __restrict__ norm, int E) {
  int e = blockIdx.x * blockDim.x + threadIdx.x;
  if (e < E) {
    int r = (int)ei[e];
    int c = (int)ei[(long)E + e];
    norm[e] = dinv[r] * dinv[c];
  }
}

// ---------------- hop propagation (SpMM) ----------------

// Write src into section 0 of Xcat[N,512]; zero sections 1..3 for atomic accumulation.
__global__ void init_xcat(const float* __restrict__ src, float* __restrict__ xcat, long n128) {
  long i = (long)blockIdx.x * blockDim.x + threadIdx.x;
  if (i < n128) {
    long node = i >> 7;
    int  f    = (int)(i & 127);
    float v = src[i];
    float* row = xcat + node * KCAT;
    row[f]       = v;
    row[128 + f] = 0.0f;
    row[256 + f] = 0.0f;
    row[384 + f] = 0.0f;
  }
}

// dst[col,:] += norm[e] * src[row,:]; one wave per edge, float4 per lane.
__global__ void spmm_hop(const long long* __restrict__ ei, const float* __restrict__ norm,
                         const float* __restrict__ src, float* __restrict__ dst, int E) {
  long tid = (long)blockIdx.x * blockDim.x + threadIdx.x;
  long e = tid >> 5;
  if (e >= E) return;
  int g = (int)(tid & 31);               // 4-float feature group
  int r = (int)ei[e];
  int c = (int)ei[(long)E + e];
  float w = norm[e];
  float4 v = *((const float4*)(src + (long)r * KCAT) + g);
  float* d = dst + (long)c * KCAT + g * 4;
  atomicAdd(d + 0, w * v.x);
  atomicAdd(d + 1, w * v.y);
  atomicAdd(d + 2, w * v.z);
  atomicAdd(d + 3, w * v.w);
}

// ---------------- fp32 WMMA GEMM: out[N,128] = A[N,512] @ W[128,512]^T + bias ----------------

__global__ void __launch_bounds__(256)
gemm_wmma_f32(const float* __restrict__ A, const float* __restrict__ W,
              const float* __restrict__ bias, float* __restrict__ out, int relu) {
  int lane = threadIdx.x & 31;
  int wave = threadIdx.x >> 5;           // ntile 0..7 (N = 128 = 8 * 16)
  int lo = lane & 15;
  int hi = lane >> 4;
  long m0 = (long)blockIdx.x * 16;

  // A lane view: row m0+lo, K = k + 2*hi (+0,+1)   [16x4 f32 A layout]
  const float* arow = A + (m0 + lo) * KCAT + 2 * hi;
  // B = W^T (4x16 K x N tile): lane lo gives N, so read W row n, cols k+2*hi (+0,+1)
  const float* brow = W + (long)(wave * 16 + lo) * KCAT + 2 * hi;

  v8f c = {};
#pragma unroll 4
  for (int k = 0; k < KCAT; k += 4) {
    v2f a = *(const v2f*)(arow + k);
    v2f b = *(const v2f*)(brow + k);
    // v_wmma_f32_16x16x4_f32: (neg_a, A, neg_b, B, c_mod, C, reuse_a, reuse_b)
    c = __builtin_amdgcn_wmma_f32_16x16x4_f32(false, a, false, b, (short)0, c, false, false);
  }

  float bb = bias[wave * 16 + lo];
  long outbase = (m0 + 8 * hi) * CH + wave * 16 + lo;
#pragma unroll
  for (int r = 0; r < 8; ++r) {
    float v = c[r] + bb;
    if (relu) v = fmaxf(v, 0.0f);
    out[outbase + (long)r * CH] = v;
  }
}

// ---------------- driver ----------------

extern "C" void kernel_launch(void* const* d_in, const int* in_sizes, int n_in,
                              void* d_out, int out_size, void* d_ws, size_t ws_size,
                              hipStream_t stream) {
  const float*     x  = (const float*)d_in[0];
  const long long* ei = (const long long*)d_in[1];   // int64 [2, E] flat: rows then cols
  const float*     W1 = (const float*)d_in[2];
  const float*     b1 = (const float*)d_in[3];
  const float*     W2 = (const float*)d_in[4];
  const float*     b2 = (const float*)d_in[5];
  float* out = (float*)d_out;

  int E = in_sizes[1] / 2;
  int N = in_sizes[0] / CH;

  float* ws   = (float*)d_ws;
  float* norm = ws;                       // E floats
  float* dinv = norm + E;                 // N floats (deg, then rsqrt in place)
  float* xcat = dinv + N;                 // N*512 floats
  float* H    = xcat + (long)N * KCAT;    // N*128 floats

  const int T = 256;

  // degree / symmetric norm (shared by both layers)
  zero_f32 <<<(N + T - 1) / T, T, 0, stream>>>(dinv, N);
  deg_kernel<<<(E + T - 1) / T, T, 0, stream>>>(ei, dinv, E);
  dinv_kernel<<<(N + T - 1) / T, T, 0, stream>>>(dinv, N);
  norm_kernel<<<(E + T - 1) / T, T, 0, stream>>>(ei, dinv, norm, E);

  long n128  = (long)N * CH;
  long spmmT = (long)E * 32;
  int  gspmm = (int)((spmmT + T - 1) / T);
  int  ginit = (int)((n128 + T - 1) / T);

  // ---- layer 1: hops over x, then Linear+ReLU -> H
  init_xcat<<<ginit, T, 0, stream>>>(x, xcat, n128);
  for (int k = 1; k <= 3; ++k)
    spmm_hop<<<gspmm, T, 0, stream>>>(ei, norm, xcat + (k - 1) * CH, xcat + k * CH, E);
  gemm_wmma_f32<<<N / 16, 256, 0, stream>>>(xcat, W1, b1, H, 1);

  // ---- layer 2: hops over H, then Linear -> out
  init_xcat<<<ginit, T, 0, stream>>>(H, xcat, n128);
  for (int k = 1; k <= 3; ++k)
    spmm_hop<<<gspmm, T, 0, stream>>>(ei, norm, xcat + (k - 1) * CH, xcat + k * CH, E);
  gemm_wmma_f32<<<N / 16, 256, 0, stream>>>(xcat, W2, b2, out, 0);
}